// Multi_Fidelity_Loss2_18829136626194
// MI455X (gfx1250) — compile-verified
//
#include <hip/hip_runtime.h>
#include <stdint.h>

// Fused masked fidelity loss on MI455X (gfx1250).
// Bandwidth-bound (1 GiB @ 23.3 TB/s ~= 46 us floor). Streaming via CDNA5
// async global->LDS (ASYNCcnt) with per-wave private staging (no barriers in
// the hot loop), progressive s_wait_asynccnt pipelining, deterministic
// two-stage reduction.

#define B_ROWS 16384
#define C_COLS 8192
#define NWAVES 8
#define CHUNK_COLS 128                     // 32 lanes * 4 floats (B128 per lane)
#define CHUNKS 8                           // per wave
#define COLS_PER_WAVE (CHUNKS * CHUNK_COLS) // 1024
#define EPS 1e-12f

// CDNA5 async DMA: copies 16 bytes from global base saddr plus offset vaddr to the LDS address in vdst. Tracked by ASYNCcnt.
__device__ __forceinline__ void async_load_b128(unsigned lds_byte_off,
                                                unsigned global_byte_off,
                                                const float* base) {
  asm volatile("global_load_async_to_lds_b128 %0, %1, %2"
               :
               : "v"(lds_byte_off), "v"(global_byte_off), "s"(base)
               : "memory");
}

__global__ void __launch_bounds__(256) fidelity_row_kernel(
    const float* __restrict__ p, const float* __restrict__ g,
    float* __restrict__ row_out) {
  // 64 KB: 8 waves x (p + g) x 8 chunks x 128 floats. Per-wave private regions.
  __shared__ __align__(16) float s_p[NWAVES][CHUNKS][CHUNK_COLS];
  __shared__ __align__(16) float s_g[NWAVES][CHUNKS][CHUNK_COLS];

  const int tid  = (int)threadIdx.x;
  const int wave = tid >> 5;
  const int lane = tid & 31;
  const int row  = (int)blockIdx.x;

  // Global byte offset of this lane's first element (fits in 31 bits).
  const unsigned gbase =
      (unsigned)(row * C_COLS + wave * COLS_PER_WAVE + lane * 4) * 4u;

  // LDS byte offsets: truncate flat address (LDS aperture is in bits [63:32]).
  const unsigned lpb = (unsigned)(uintptr_t)(&s_p[wave][0][lane * 4]);
  const unsigned lgb = (unsigned)(uintptr_t)(&s_g[wave][0][lane * 4]);

  // Queue all 16 async B128 transfers for this wave (deep MLP, zero VGPR cost).
#pragma unroll
  for (int ch = 0; ch < CHUNKS; ++ch) {
    const unsigned d = (unsigned)(ch * CHUNK_COLS * 4);
    async_load_b128(lpb + d, gbase + d, p);
    async_load_b128(lgb + d, gbase + d, g);
  }

  float sum = 0.0f, cnt = 0.0f;
  auto acc = [&](float pe, float ge) {
    const float a = __builtin_amdgcn_sqrtf(fmaf(pe, ge, EPS));
    const float b = __builtin_amdgcn_sqrtf(fmaf(1.0f - pe, 1.0f - ge, EPS));
    const float l = 1.0f - a - b;
    const float m = (ge != 0.0f) ? 1.0f : 0.0f;
    sum = fmaf(m, l, sum);
    cnt += m;
  };

  // Consume chunk CH once the first 2*(CH+1) of this wave's 16 in-order async
  // loads have retired: outstanding <= 14 - 2*CH.
#define FID_STEP(CH, WN)                                                       \
  {                                                                            \
    asm volatile("s_wait_asynccnt " #WN ::: "memory");                         \
    const float4 pv = *(const float4*)(&s_p[wave][CH][lane * 4]);              \
    const float4 gv = *(const float4*)(&s_g[wave][CH][lane * 4]);              \
    acc(pv.x, gv.x);                                                           \
    acc(pv.y, gv.y);                                                           \
    acc(pv.z, gv.z);                                                           \
    acc(pv.w, gv.w);                                                           \
  }

  FID_STEP(0, 14)
  FID_STEP(1, 12)
  FID_STEP(2, 10)
  FID_STEP(3, 8)
  FID_STEP(4, 6)
  FID_STEP(5, 4)
  FID_STEP(6, 2)
  FID_STEP(7, 0)
#undef FID_STEP

  // Block reduction (reuse staging LDS; ASYNCcnt is 0 for every wave here).
  __syncthreads();
  float* red = &s_p[0][0][0];  // [0..255]=sum, [256..511]=cnt
  red[tid] = sum;
  red[256 + tid] = cnt;
  __syncthreads();
#pragma unroll
  for (int s = 128; s > 0; s >>= 1) {
    if (tid < s) {
      red[tid] += red[tid + s];
      red[256 + tid] += red[256 + tid + s];
    }
    __syncthreads();
  }
  if (tid == 0) {
    const float c = red[256];
    row_out[row] = (c > 0.0f) ? (red[0] / c) : 0.0f;
  }
}

__global__ void __launch_bounds__(256) final_reduce_kernel(
    const float* __restrict__ row_loss, float* __restrict__ out) {
  __shared__ float sm[256];
  const int tid = (int)threadIdx.x;
  float s = 0.0f;
#pragma unroll 8
  for (int i = tid; i < B_ROWS; i += 256) s += row_loss[i];
  sm[tid] = s;
  __syncthreads();
#pragma unroll
  for (int k = 128; k > 0; k >>= 1) {
    if (tid < k) sm[tid] += sm[tid + k];
    __syncthreads();
  }
  if (tid == 0) out[0] = sm[0] * (1.0f / (float)B_ROWS);
}

extern "C" void kernel_launch(void* const* d_in, const int* in_sizes, int n_in,
                              void* d_out, int out_size, void* d_ws,
                              size_t ws_size, hipStream_t stream) {
  (void)in_sizes; (void)n_in; (void)out_size; (void)ws_size;
  const float* p = (const float*)d_in[0];
  const float* g = (const float*)d_in[1];
  float* row_loss = (float*)d_ws;   // 16384 floats = 64 KB scratch
  float* out = (float*)d_out;

  fidelity_row_kernel<<<dim3(B_ROWS), dim3(256), 0, stream>>>(p, g, row_loss);
  final_reduce_kernel<<<dim3(1), dim3(256), 0, stream>>>(row_loss, out);
}